// get_model_48361331753047
// MI455X (gfx1250) — compile-verified
//
#include <hip/hip_runtime.h>
#include <hip/hip_bf16.h>

typedef float v2f __attribute__((ext_vector_type(2)));
typedef float v8f __attribute__((ext_vector_type(8)));

#define BN_INV 0.99999500003750f  // 1/sqrt(1+1e-5)

__device__ __forceinline__ v8f wmma_f32_k4(v2f a, v2f b, v8f c) {
  // D = A(16x4) * B(4x16) + C, exact fp32 matrix op on CDNA5
  return __builtin_amdgcn_wmma_f32_16x16x4_f32(false, a, false, b, (short)0, c, false, false);
}

// ---------------------------------------------------------------- split (B,6,N) -> pts/norm (B,N,3)
__global__ void split_xyz_kernel(const float* __restrict__ xyz, float* __restrict__ pts,
                                 float* __restrict__ nrm, int B, int N) {
  int t = blockIdx.x * blockDim.x + threadIdx.x;
  int total = B * N * 3;
  if (t >= total) return;
  int c = t % 3;
  int bn = t / 3;
  int n = bn % N;
  int b = bn / N;
  pts[t] = xyz[((size_t)b * 6 + c) * N + n];
  nrm[t] = xyz[((size_t)b * 6 + 3 + c) * N + n];
}

// ---------------------------------------------------------------- farthest point sampling
__global__ __launch_bounds__(256) void fps_kernel(const float* __restrict__ xyz, int N, int npoint,
                                                  int* __restrict__ outIdx, float* __restrict__ newXyz) {
  __shared__ float rv[256];
  __shared__ int ri[256];
  __shared__ float sc[3];
  int b = blockIdx.x, tid = threadIdx.x;
  int cnt = N >> 8;  // points per thread (16 for N=4096, 2 for N=512)
  float dist[16];
  for (int i = 0; i < cnt; ++i) dist[i] = 1e10f;
  int base = tid * cnt;
  int far = 0;
  for (int it = 0; it < npoint; ++it) {
    if (tid == 0) {
      outIdx[b * npoint + it] = far;
      const float* p = xyz + ((size_t)b * N + far) * 3;
      sc[0] = p[0]; sc[1] = p[1]; sc[2] = p[2];
      float* o = newXyz + ((size_t)b * npoint + it) * 3;
      o[0] = p[0]; o[1] = p[1]; o[2] = p[2];
    }
    __syncthreads();
    float cx = sc[0], cy = sc[1], cz = sc[2];
    float bm = -1.f;
    int bi = base;
    for (int i = 0; i < cnt; ++i) {
      int p = base + i;
      const float* q = xyz + ((size_t)b * N + p) * 3;
      float dx = q[0] - cx, dy = q[1] - cy, dz = q[2] - cz;
      float d = dx * dx + dy * dy + dz * dz;
      float nd = fminf(dist[i], d);
      dist[i] = nd;
      if (nd > bm) { bm = nd; bi = p; }
    }
    rv[tid] = bm; ri[tid] = bi;
    __syncthreads();
    for (int s = 128; s > 0; s >>= 1) {
      if (tid < s) {
        float v2 = rv[tid + s];
        int i2 = ri[tid + s];
        if (v2 > rv[tid] || (v2 == rv[tid] && i2 < ri[tid])) { rv[tid] = v2; ri[tid] = i2; }
      }
      __syncthreads();
    }
    far = ri[0];
    __syncthreads();
  }
}

// ---------------------------------------------------------------- ball query (ascending-index scan)
__global__ void ball_query_kernel(const float* __restrict__ xyz, const float* __restrict__ newXyz,
                                  int N, int S, int ns, float r2, int* __restrict__ idx, int BS) {
  int t = blockIdx.x * blockDim.x + threadIdx.x;
  if (t >= BS) return;
  int b = t / S;
  const float* c = newXyz + (size_t)t * 3;
  float cx = c[0], cy = c[1], cz = c[2];
  int* o = idx + (size_t)t * ns;
  int cnt = 0;
  for (int n = 0; n < N && cnt < ns; ++n) {
    const float* p = xyz + ((size_t)b * N + n) * 3;
    float dx = p[0] - cx, dy = p[1] - cy, dz = p[2] - cz;
    float d = dx * dx + dy * dy + dz * dz;
    if (d <= r2) o[cnt++] = n;
  }
  int first = (cnt > 0) ? o[0] : 0;
  for (; cnt < ns; ++cnt) o[cnt] = first;
}

// ---------------------------------------------------------------- fused SA MLP layer (LDS -> LDS), WMMA f32
// K/Kpad are compile-time after inlining; clamp+select (no exec branches) handles the K<Kpad pad.
__device__ __forceinline__ void mlp_layer_lds(const float* __restrict__ src, int srcStride,
                                              int K, int Kpad,
                                              const float* __restrict__ W, const float* __restrict__ bias,
                                              const float* __restrict__ g, const float* __restrict__ be,
                                              float* __restrict__ dst, int dstStride, int Cout) {
  int lane = threadIdx.x & 31;
  int wave = threadIdx.x >> 5;
  int kh = lane >> 4;
  int l = lane & 15;
  int ntot = (Cout >> 4) * 2;  // 2 row tiles x Cout/16 col tiles
  for (int t = wave; t < ntot; t += 4) {
    int tm = t & 1, tn = t >> 1;
    v8f acc = {0.f, 0.f, 0.f, 0.f, 0.f, 0.f, 0.f, 0.f};
    int rowA = tm * 16 + l;
    int colW = tn * 16 + l;
    const float* srow = src + rowA * srcStride;
    const float* wrow = W + (size_t)colW * K;
    for (int k0 = 0; k0 < Kpad; k0 += 4) {
      int ka = k0 + kh * 2;
      v2f a, bv;
      a.x = srow[ka];
      a.y = srow[ka + 1];
      float w0 = wrow[ka < K ? ka : K - 1];          // unconditional load, clamped addr
      float w1 = wrow[ka + 1 < K ? ka + 1 : K - 1];
      bv.x = (ka < K) ? w0 : 0.f;                    // v_cndmask, no branch
      bv.y = (ka + 1 < K) ? w1 : 0.f;
      acc = wmma_f32_k4(a, bv, acc);
    }
    float scale = g[colW] * BN_INV, shift = be[colW], bb = bias[colW];
#pragma unroll
    for (int r = 0; r < 8; ++r) {
      int row = tm * 16 + kh * 8 + r;
      float v = scale * (acc[r] + bb) + shift;
      v = fmaxf(v, 0.f);
      dst[row * dstStride + colW] = v;
    }
  }
}

// last layer fused with group max: shuffle across halves + global atomic-max (post-relu >= 0)
__device__ __forceinline__ void mlp_layer_max(const float* __restrict__ src, int srcStride, int K,
                                              const float* __restrict__ W, const float* __restrict__ bias,
                                              const float* __restrict__ g, const float* __restrict__ be,
                                              float* __restrict__ out, int Cout) {
  int lane = threadIdx.x & 31;
  int wave = threadIdx.x >> 5;
  int kh = lane >> 4;
  int l = lane & 15;
  int ntot = (Cout >> 4) * 2;
  for (int t = wave; t < ntot; t += 4) {
    int tm = t & 1, tn = t >> 1;
    v8f acc = {0.f, 0.f, 0.f, 0.f, 0.f, 0.f, 0.f, 0.f};
    int rowA = tm * 16 + l;
    int colW = tn * 16 + l;
    const float* srow = src + rowA * srcStride;
    const float* wrow = W + (size_t)colW * K;
    for (int k0 = 0; k0 < K; k0 += 4) {
      int ka = k0 + kh * 2;
      v2f a, bv;
      a.x = srow[ka];
      a.y = srow[ka + 1];
      bv.x = wrow[ka];
      bv.y = wrow[ka + 1];
      acc = wmma_f32_k4(a, bv, acc);
    }
    float scale = g[colW] * BN_INV, shift = be[colW], bb = bias[colW];
    float m = 0.f;  // relu floor
#pragma unroll
    for (int r = 0; r < 8; ++r) {
      float v = scale * (acc[r] + bb) + shift;
      m = fmaxf(m, v);
    }
    m = fmaxf(m, __shfl_xor(m, 16, 32));  // combine rows 0-7 / 8-15 of tile
    if (lane < 16) atomicMax((int*)(out + colW), __float_as_int(m));
  }
}

// block = 32 grouped rows of one (b,s) center; gather + 3x (GEMM+BN+ReLU) + max, all in LDS
template <int CIN, int C1, int C2, int C3>
__global__ __launch_bounds__(128) void sa_mlp_kernel(
    const float* __restrict__ pts, const float* __restrict__ feat, int featDim, int Npts,
    const int* __restrict__ idx, const float* __restrict__ centers, int S, int NS,
    const float* __restrict__ W1, const float* __restrict__ b1, const float* __restrict__ g1, const float* __restrict__ be1,
    const float* __restrict__ W2, const float* __restrict__ b2, const float* __restrict__ g2, const float* __restrict__ be2,
    const float* __restrict__ W3, const float* __restrict__ b3, const float* __restrict__ g3, const float* __restrict__ be3,
    float* __restrict__ out) {
  constexpr int CINP = (CIN + 3) & ~3;
  constexpr int SIN = CINP + 1;  // odd strides -> conflict-free column access
  constexpr int SA_ = C1 + 1;
  constexpr int SB_ = C2 + 1;
  __shared__ float bIn[32 * SIN];
  __shared__ float bA[32 * SA_];
  __shared__ float bB[32 * SB_];
  int bs = blockIdx.x;   // b*S + s
  int b = bs / S;
  int rowOff = blockIdx.y * 32;
  const int* gi = idx + (size_t)bs * NS + rowOff;
  const float* ctr = centers + (size_t)bs * 3;
  for (int e = threadIdx.x; e < 32 * CINP; e += 128) {
    int j = e / CINP, c = e % CINP;
    float v = 0.f;
    if (c < CIN) {
      int p = gi[j];
      if (c < 3) v = pts[((size_t)b * Npts + p) * 3 + c] - ctr[c];
      else       v = feat[((size_t)b * Npts + p) * featDim + (c - 3)];
    }
    bIn[j * SIN + c] = v;
  }
  __syncthreads();
  mlp_layer_lds(bIn, SIN, CIN, CINP, W1, b1, g1, be1, bA, SA_, C1);
  __syncthreads();
  mlp_layer_lds(bA, SA_, C1, C1, W2, b2, g2, be2, bB, SB_, C2);
  __syncthreads();
  mlp_layer_max(bB, SB_, C2, W3, b3, g3, be3, out + (size_t)bs * C3, C3);
}

// ---------------------------------------------------------------- generic WMMA GEMM: Y = act(bn(X @ W^T + b))
// Main K-loop is branch-free: OOB rows/cols use clamped pointers (garbage lands only in
// rows/cols the epilogue never stores). Only the K%4 tail is value-predicated.
__global__ __launch_bounds__(128) void gemm_bn_relu_kernel(
    const float* __restrict__ X, int ldx, const float* __restrict__ W,
    const float* __restrict__ bias, const float* __restrict__ g, const float* __restrict__ be,
    float* __restrict__ Y, int ldy, int colOff, int M, int N, int K, int doRelu) {
  int wave = threadIdx.x >> 5;
  int lane = threadIdx.x & 31;
  int tilesN = (N + 15) >> 4;
  int tilesM = (M + 15) >> 4;
  int t = blockIdx.x * 4 + wave;
  if (t >= tilesM * tilesN) return;  // wave-uniform
  int tn = t % tilesN, tm = t / tilesN;
  int kh = lane >> 4, l = lane & 15;
  int row = tm * 16 + l;
  int col = tn * 16 + l;
  int rowc = row < M ? row : M - 1;  // clamp: always in-bounds
  int colc = col < N ? col : N - 1;
  const float* xr = X + (size_t)rowc * ldx;
  const float* wr = W + (size_t)colc * K;
  v8f acc = {0.f, 0.f, 0.f, 0.f, 0.f, 0.f, 0.f, 0.f};
  int Kmain = K & ~3;
#pragma unroll 2
  for (int k0 = 0; k0 < Kmain; k0 += 4) {
    int ka = k0 + kh * 2;
    v2f a, bv;
    a.x = xr[ka];
    a.y = xr[ka + 1];
    bv.x = wr[ka];
    bv.y = wr[ka + 1];
    acc = wmma_f32_k4(a, bv, acc);
  }
  if (K & 3) {  // tail (only K=259 path): clamp addresses, select values -> no branches
    int ka = Kmain + kh * 2;
    int ka0 = ka < K ? ka : K - 1;
    int ka1 = ka + 1 < K ? ka + 1 : K - 1;
    float x0 = xr[ka0], x1 = xr[ka1];
    float w0 = wr[ka0], w1 = wr[ka1];
    v2f a, bv;
    a.x = (ka < K) ? x0 : 0.f;
    a.y = (ka + 1 < K) ? x1 : 0.f;
    bv.x = (ka < K) ? w0 : 0.f;
    bv.y = (ka + 1 < K) ? w1 : 0.f;
    acc = wmma_f32_k4(a, bv, acc);
  }
  float scale = g ? g[colc] * BN_INV : 1.f;
  float shift = be ? be[colc] : 0.f;
  float bb = bias ? bias[colc] : 0.f;
#pragma unroll
  for (int r = 0; r < 8; ++r) {
    int ro = tm * 16 + kh * 8 + r;
    if (ro < M && col < N) {
      float v = scale * (acc[r] + bb) + shift;
      if (doRelu) v = fmaxf(v, 0.f);
      Y[(size_t)ro * ldy + colOff + col] = v;
    }
  }
}

// ---------------------------------------------------------------- SA3 helpers
__global__ void build_x3_kernel(const float* __restrict__ l2xyz, const float* __restrict__ l2pts,
                                float* __restrict__ X3, int total) {
  int t = blockIdx.x * blockDim.x + threadIdx.x;
  if (t >= total) return;
  int c = t % 259;
  int r = t / 259;
  X3[t] = (c < 3) ? l2xyz[(size_t)r * 3 + c] : l2pts[(size_t)r * 256 + (c - 3)];
}

__global__ void max128_kernel(const float* __restrict__ Y, float* __restrict__ l3,
                              float* __restrict__ outTail, int total) {
  int t = blockIdx.x * blockDim.x + threadIdx.x;
  if (t >= total) return;
  int b = t >> 10, c = t & 1023;
  float m = -3.4e38f;
  const float* p = Y + ((size_t)b * 128) * 1024 + c;
  for (int r = 0; r < 128; ++r) m = fmaxf(m, p[(size_t)r * 1024]);
  l3[t] = m;
  outTail[t] = m;  // l3_points output (B,1024,1) == flat b*1024+c
}

// ---------------------------------------------------------------- host orchestration
extern "C" void kernel_launch(void* const* d_in, const int* in_sizes, int n_in,
                              void* d_out, int out_size, void* d_ws, size_t ws_size,
                              hipStream_t stream) {
  (void)in_sizes; (void)out_size; (void)ws_size;
  const int B = 16, N = 4096;
  const float* xyz = (const float*)d_in[0];
  const float* action = (const float*)d_in[1];
  const float* P[64];
  for (int i = 0; i < n_in && i < 64; ++i) P[i] = (const float*)d_in[i];

  float* f = (float*)d_ws;
  size_t off = 0;
  auto alloc = [&](size_t n) { float* p = f + off; off += n; return p; };
  float* pts  = alloc((size_t)B * N * 3);
  float* nrm  = alloc((size_t)B * N * 3);
  float* nx1  = alloc((size_t)B * 512 * 3);
  float* l1p  = alloc((size_t)B * 512 * 128);
  float* nx2  = alloc((size_t)B * 128 * 3);
  float* l2p  = alloc((size_t)B * 128 * 256);
  float* X3   = alloc((size_t)B * 128 * 259);
  float* Y3a  = alloc((size_t)B * 128 * 256);
  float* Y3b  = alloc((size_t)B * 128 * 512);
  float* Y3c  = alloc((size_t)B * 128 * 1024);
  float* l3   = alloc((size_t)B * 1024);
  float* fx1  = alloc((size_t)B * 512);
  float* c288 = alloc((size_t)B * 288);
  float* a1   = alloc((size_t)B * 64);
  float* h1   = alloc((size_t)B * 64);
  float* h2   = alloc((size_t)B * 32);
  int* fpsi1 = (int*)(f + off);
  int* idx1  = fpsi1 + B * 512;
  int* fpsi2 = idx1 + (size_t)B * 512 * 32;
  int* idx2  = fpsi2 + B * 128;

  { int tot = B * N * 3;
    split_xyz_kernel<<<(tot + 255) / 256, 256, 0, stream>>>(xyz, pts, nrm, B, N); }

  // ---- SA1: fps 512, ball 0.2/32, MLP 6->64->64->128, max over 32
  fps_kernel<<<B, 256, 0, stream>>>(pts, N, 512, fpsi1, nx1);
  { int bs = B * 512;
    ball_query_kernel<<<(bs + 127) / 128, 128, 0, stream>>>(pts, nx1, N, 512, 32, 0.04f, idx1, bs); }
  hipMemsetAsync(l1p, 0, (size_t)B * 512 * 128 * sizeof(float), stream);
  sa_mlp_kernel<6, 64, 64, 128><<<dim3(B * 512, 1), 128, 0, stream>>>(
      pts, nrm, 3, N, idx1, nx1, 512, 32,
      P[2], P[3], P[4], P[5], P[6], P[7], P[8], P[9], P[10], P[11], P[12], P[13], l1p);

  // ---- SA2: fps 128, ball 0.4/64, MLP 131->128->128->256, max over 64 (2 row-blocks of 32)
  fps_kernel<<<B, 256, 0, stream>>>(nx1, 512, 128, fpsi2, nx2);
  { int bs = B * 128;
    ball_query_kernel<<<(bs + 127) / 128, 128, 0, stream>>>(nx1, nx2, 512, 128, 64, 0.16f, idx2, bs); }
  hipMemsetAsync(l2p, 0, (size_t)B * 128 * 256 * sizeof(float), stream);
  sa_mlp_kernel<131, 128, 128, 256><<<dim3(B * 128, 2), 128, 0, stream>>>(
      nx1, l1p, 128, 512, idx2, nx2, 128, 64,
      P[14], P[15], P[16], P[17], P[18], P[19], P[20], P[21], P[22], P[23], P[24], P[25], l2p);

  // ---- SA3 (group all): X3 = [l2_xyz | l2_points] (2048 x 259), MLP 259->256->512->1024, max over 128
  { int tot = B * 128 * 259;
    build_x3_kernel<<<(tot + 255) / 256, 256, 0, stream>>>(nx2, l2p, X3, tot); }

  auto gemm = [&](const float* Xp, int ldx, const float* Wp, const float* bp, const float* gp,
                  const float* bep, float* Yp, int ldy, int colOff, int M, int Nn, int K, int relu) {
    int tiles = ((M + 15) / 16) * ((Nn + 15) / 16);
    gemm_bn_relu_kernel<<<(tiles + 3) / 4, 128, 0, stream>>>(Xp, ldx, Wp, bp, gp, bep,
                                                             Yp, ldy, colOff, M, Nn, K, relu);
  };
  gemm(X3, 259, P[26], P[27], P[28], P[29], Y3a, 256, 0, B * 128, 256, 259, 1);
  gemm(Y3a, 256, P[30], P[31], P[32], P[33], Y3b, 512, 0, B * 128, 512, 256, 1);
  gemm(Y3b, 512, P[34], P[35], P[36], P[37], Y3c, 1024, 0, B * 128, 1024, 512, 1);

  float* outF = (float*)d_out;
  { int tot = B * 1024;
    max128_kernel<<<(tot + 255) / 256, 256, 0, stream>>>(Y3c, l3, outF + B * 40, tot); }

  // ---- FC heads
  gemm(l3, 1024, P[38], P[39], P[40], P[41], fx1, 512, 0, B, 512, 1024, 1);   // fc1+bn1
  gemm(fx1, 512, P[42], P[43], P[44], P[45], c288, 288, 0, B, 256, 512, 1);   // fc2+bn2 -> concat[:,0:256]
  gemm(action, 16, P[46], P[47], P[48], P[49], a1, 64, 0, B, 64, 16, 1);      // afc1+abn1
  gemm(a1, 64, P[50], P[51], P[52], P[53], c288, 288, 256, B, 32, 64, 1);     // afc2+abn2 -> concat[:,256:288]
  gemm(c288, 288, P[54], P[55], P[56], P[57], h1, 64, 0, B, 64, 288, 1);      // hfc1+hbn1
  gemm(h1, 64, P[58], P[59], P[60], P[61], h2, 32, 0, B, 32, 64, 1);          // hfc2+hbn2
  gemm(h2, 32, P[62], P[63], nullptr, nullptr, outF, 40, 0, B, 40, 32, 0);    // hfc3 -> c output
}